// rel_infer_27144193310749
// MI455X (gfx1250) — compile-verified
//
#include <hip/hip_runtime.h>
#include <math.h>

#define NROI   128
#define NCLS   1001
#define NREL   51
#define TOPK   10
#define NK     (NROI * TOPK)      // 1280
#define ROUNDS 3
#define MIXF   10000.0f

typedef float v2f __attribute__((ext_vector_type(2)));
typedef float v8f __attribute__((ext_vector_type(8)));

// ---------------------------------------------------------------------------
// Kernel 1: per-row top-K (value desc, ties -> lower index, matching lax.top_k)
// one block per row; row cached in LDS; 10 argmax reductions.
// ---------------------------------------------------------------------------
__global__ void k_topk(const float* __restrict__ roi,
                       float* __restrict__ tsc, int* __restrict__ lab,
                       float* __restrict__ unary) {
  __shared__ float row[NCLS];
  __shared__ float sval[256];
  __shared__ int   sidx[256];
  const int a = blockIdx.x;
  const int t = threadIdx.x;
  for (int c = t; c < NCLS; c += 256) row[c] = roi[a * NCLS + c];
  __syncthreads();
  for (int k = 0; k < TOPK; ++k) {
    float bv = -INFINITY; int bi = 0x7fffffff;
    for (int c = t; c < NCLS; c += 256) {
      float v = row[c];
      if (v > bv || (v == bv && c < bi)) { bv = v; bi = c; }
    }
    sval[t] = bv; sidx[t] = bi;
    __syncthreads();
    for (int s = 128; s > 0; s >>= 1) {
      if (t < s) {
        float ov = sval[t + s]; int oi = sidx[t + s];
        if (ov > sval[t] || (ov == sval[t] && oi < sidx[t])) {
          sval[t] = ov; sidx[t] = oi;
        }
      }
      __syncthreads();
    }
    if (t == 0) {
      tsc[a * TOPK + k]   = sval[0];
      lab[a * TOPK + k]   = sidx[0];
      unary[a * TOPK + k] = logf(sval[0]);
      row[sidx[0]] = -INFINITY;
    }
    __syncthreads();
  }
}

// ---------------------------------------------------------------------------
// Kernel 2: build symmetric log-potential BT (1280x1280 f32, L2-resident).
// One block per (a,b) pair, a<=b only; writes both tiles.
// BT[(a,k),(b,l)] = log(0.5*(dot(relmat[la,lb,:], e^{rel[a,b,:]}) +
//                            dot(relmat[lb,la,:], e^{rel[b,a,:]})))
// diagonal blocks (a==b) forced to 0.
// ---------------------------------------------------------------------------
__global__ void k_bt(const float* __restrict__ rel_scores,
                     const float* __restrict__ relmat,
                     const int* __restrict__ lab,
                     float* __restrict__ BT) {
  const int pair = blockIdx.x;
  const int a = pair / NROI, b = pair % NROI;
  if (b < a) return;
  __shared__ float e_ab[NREL], e_ba[NREL];
  __shared__ int   la[TOPK], lb[TOPK];
  const int t = threadIdx.x;
  if (t < NREL) {
    e_ab[t] = expf(rel_scores[(a * NROI + b) * NREL + t]);
    e_ba[t] = expf(rel_scores[(b * NROI + a) * NREL + t]);
  }
  if (t < TOPK) { la[t] = lab[a * TOPK + t]; lb[t] = lab[b * TOPK + t]; }
  __syncthreads();
  if (t < TOPK * TOPK) {
    const int k = t / TOPK, l = t % TOPK;
    float v;
    if (a == b) {
      v = 0.0f;   // eye(n) mask over whole diagonal block
    } else {
      const float* r1 = relmat + ((size_t)la[k] * NCLS + lb[l]) * NREL;
      const float* r2 = relmat + ((size_t)lb[l] * NCLS + la[k]) * NREL;
      float d1 = 0.0f, d2 = 0.0f;
      #pragma unroll 3
      for (int r = 0; r < NREL; ++r) {
        d1 = fmaf(r1[r], e_ab[r], d1);
        d2 = fmaf(r2[r], e_ba[r], d2);
      }
      v = logf(0.5f * (d1 + d2));
    }
    BT[(size_t)(a * TOPK + k) * NK + (b * TOPK + l)] = v;
    BT[(size_t)(b * TOPK + l) * NK + (a * TOPK + k)] = v;
  }
}

// ---------------------------------------------------------------------------
// Kernel 3: Q init
// ---------------------------------------------------------------------------
__global__ void k_init_q(float* __restrict__ Q) {
  int i = blockIdx.x * 256 + threadIdx.x;
  if (i < NK) Q[i] = 1.0f / (float)TOPK;
}

// ---------------------------------------------------------------------------
// Kernel 4: msg = BT @ Q via V_WMMA_F32_16X16X4_F32 (exact f32 GEMV).
// One wave (32 lanes) per 16 output rows. Q is replicated across all 16
// B-columns, making the result independent of the N-lane mapping.
// A 16x4 f32 layout: lanes 0-15 -> M=lane, K={0,1}; lanes 16-31 -> K={2,3}.
// C/D layout: VGPR v -> M = v + 8*(lane/16), N = lane%16.
// ---------------------------------------------------------------------------
__global__ void k_msg_wmma(const float* __restrict__ BT,
                           const float* __restrict__ Q,
                           float* __restrict__ msg) {
  const int m    = blockIdx.x * 16;
  const int L    = threadIdx.x;        // 0..31, full wave32
  const int row  = m + (L & 15);
  const int koff = (L >> 4) * 2;       // 0 or 2
  const float* Arow = BT + (size_t)row * NK;
  v8f acc = {};
  for (int kk = 0; kk < NK; kk += 4) {
    v2f Aa, Bb;
    Aa.x = Arow[kk + koff];
    Aa.y = Arow[kk + koff + 1];
    Bb.x = Q[kk + koff];               // replicated over all N columns
    Bb.y = Q[kk + koff + 1];
    acc = __builtin_amdgcn_wmma_f32_16x16x4_f32(
        /*neg_a=*/false, Aa, /*neg_b=*/false, Bb,
        /*c_mod=*/(short)0, acc, /*reuse_a=*/false, /*reuse_b=*/false);
  }
  if ((L & 15) == 0) {                 // N==0 column holders: lanes 0 and 16
    const int off = (L >> 4) * 8;
    #pragma unroll
    for (int v = 0; v < 8; ++v) msg[m + off + v] = acc[v];
  }
}

// ---------------------------------------------------------------------------
// Kernel 5: Q = softmax(msg + unary) per row of 10
// ---------------------------------------------------------------------------
__global__ void k_softmax(const float* __restrict__ msg,
                          const float* __restrict__ unary,
                          float* __restrict__ Q) {
  const int a = threadIdx.x;           // one block of 128 threads
  if (a < NROI) {
    float x[TOPK]; float mx = -INFINITY;
    #pragma unroll
    for (int k = 0; k < TOPK; ++k) {
      x[k] = msg[a * TOPK + k] + unary[a * TOPK + k];  // W_BIN = 1
      mx = fmaxf(mx, x[k]);
    }
    float s = 0.0f;
    #pragma unroll
    for (int k = 0; k < TOPK; ++k) { x[k] = expf(x[k] - mx); s += x[k]; }
    const float inv = 1.0f / s;
    #pragma unroll
    for (int k = 0; k < TOPK; ++k) Q[a * TOPK + k] = x[k] * inv;
  }
}

// ---------------------------------------------------------------------------
// Kernels 6/7: output = roi/(1+MIX), then += MIX*Q/(1+MIX) at topk labels
// (labels are distinct within a row, so the RMW is race-free)
// ---------------------------------------------------------------------------
__global__ void k_scale(const float* __restrict__ roi, float* __restrict__ out) {
  int i = blockIdx.x * 256 + threadIdx.x;
  if (i < NROI * NCLS) out[i] = roi[i] * (1.0f / (1.0f + MIXF));
}

__global__ void k_scatter(const int* __restrict__ lab,
                          const float* __restrict__ Q,
                          float* __restrict__ out) {
  int t = blockIdx.x * 256 + threadIdx.x;
  if (t < NK) {
    int a = t / TOPK;
    int c = lab[t];
    out[a * NCLS + c] += (MIXF / (1.0f + MIXF)) * Q[t];
  }
}

// ---------------------------------------------------------------------------
extern "C" void kernel_launch(void* const* d_in, const int* in_sizes, int n_in,
                              void* d_out, int out_size, void* d_ws, size_t ws_size,
                              hipStream_t stream) {
  (void)in_sizes; (void)n_in; (void)out_size; (void)ws_size;
  const float* roi    = (const float*)d_in[0];   // (128,1001)
  const float* rel    = (const float*)d_in[1];   // (16384,51)
  const float* relmat = (const float*)d_in[2];   // (1001,1001,51)
  float* out = (float*)d_out;

  char* ws = (char*)d_ws;
  float* BT    = (float*)ws;                         // 1280*1280*4 = 6,553,600 B
  float* msg   = (float*)(ws + (size_t)NK * NK * 4); // 1280 f
  float* Q     = msg + NK;
  float* unary = Q + NK;
  float* tsc   = unary + NK;
  int*   lab   = (int*)(tsc + NK);

  k_topk<<<NROI, 256, 0, stream>>>(roi, tsc, lab, unary);
  k_bt<<<NROI * NROI, 128, 0, stream>>>(rel, relmat, lab, BT);
  k_init_q<<<(NK + 255) / 256, 256, 0, stream>>>(Q);
  for (int r = 0; r < ROUNDS; ++r) {
    k_msg_wmma<<<NK / 16, 32, 0, stream>>>(BT, Q, msg);
    k_softmax<<<1, 128, 0, stream>>>(msg, unary, Q);
  }
  k_scale<<<(NROI * NCLS + 255) / 256, 256, 0, stream>>>(roi, out);
  k_scatter<<<(NK + 255) / 256, 256, 0, stream>>>(lab, Q, out);
}